// SensorGAT_82575041232964
// MI455X (gfx1250) — compile-verified
//
#include <hip/hip_runtime.h>
#include <hip/hip_bf16.h>

typedef float v2f __attribute__((ext_vector_type(2)));
typedef float v8f __attribute__((ext_vector_type(8)));

#define LRELU_SLOPE 0.2f
#define BN_EPS 1e-5f

// ---------------------------------------------------------------------------
// float atomic max via sign-split int/uint atomics (always available, lowers
// to global_atomic_max_i32 / global_atomic_min_u32)
// ---------------------------------------------------------------------------
__device__ __forceinline__ void atomicMaxF32(float* addr, float val) {
    if (val >= 0.0f) {
        atomicMax((int*)addr, __float_as_int(val));
    } else {
        atomicMin((unsigned int*)addr, __float_as_uint(val));
    }
}

// ---------------------------------------------------------------------------
// OUT[N,64] = IN[N,FIN] @ W[64,FIN]^T  via V_WMMA_F32_16X16X4_F32 (fp32 exact)
// block = 256 threads = 8 waves; block tile = 32 rows x 64 cols
// FIN is compile-time: k-loop fully unrolled, guards are value-selects
// (v_cndmask) not EXEC branches -> WMMA runs with EXEC all-ones, loads are
// unconditional b64s for FIN=64. Store epilogue: one uniform-ish branch,
// fast path = 8 unguarded stores (always taken when nrows % 32 == 0).
// A frag (16x4 f32): lanes 0-15: M=lane,   {K=0,K=1}; lanes 16-31: M=lane-16, {K=2,K=3}
// B frag (4x16 f32): lanes 0-15: N=lane,   {K=0,K=1}; lanes 16-31: N=lane-16, {K=2,K=3}
// C (16x16 f32):     vgpr v: lanes 0-15 -> M=v, N=lane; lanes 16-31 -> M=v+8
// ---------------------------------------------------------------------------
template <int FIN>
__global__ void gemm64_wmma(const float* __restrict__ in, const float* __restrict__ w,
                            float* __restrict__ out, int nrows) {
    const int wave = threadIdx.x >> 5;
    const int lane = threadIdx.x & 31;
    const int row0 = blockIdx.x * 32 + (wave >> 2) * 16;
    const int col0 = (wave & 3) * 16;
    const int lmod = lane & 15;
    const bool hi = lane >= 16;

    int aRow = row0 + lmod;
    aRow = (aRow < nrows) ? aRow : (nrows - 1);          // branch-free clamp
    const float* __restrict__ arow = in + (size_t)aRow * FIN;
    const float* __restrict__ brow = w + (size_t)(col0 + lmod) * FIN;

    v8f acc = {0.f, 0.f, 0.f, 0.f, 0.f, 0.f, 0.f, 0.f};
    constexpr int KSTEPS = (FIN + 3) / 4;
#pragma unroll
    for (int ks = 0; ks < KSTEPS; ++ks) {
        const int k0 = ks * 4 + (hi ? 2 : 0);
        const int ka = (k0     < FIN) ? k0       : (FIN - 1);   // clamped, always valid
        const int kb = (k0 + 1 < FIN) ? (k0 + 1) : (FIN - 1);
        const float a0 = arow[ka];
        const float a1 = arow[kb];
        const float b0 = brow[ka];
        const float b1 = brow[kb];
        v2f a, b;
        a.x = (k0     < FIN) ? a0 : 0.f;    // folds away for FIN=64
        a.y = (k0 + 1 < FIN) ? a1 : 0.f;
        b.x = (k0     < FIN) ? b0 : 0.f;
        b.y = (k0 + 1 < FIN) ? b1 : 0.f;
        acc = __builtin_amdgcn_wmma_f32_16x16x4_f32(
            /*neg_a=*/false, a, /*neg_b=*/false, b,
            /*c_mod=*/(short)0, acc, /*reuse_a=*/false, /*reuse_b=*/false);
    }

    const int mbase = row0 + (hi ? 8 : 0);
    const int ncol = col0 + lmod;
    float* __restrict__ obase = out + (size_t)mbase * 64 + ncol;
    if (mbase + 7 < nrows) {
        // fast path: all 8 rows in range (always, when nrows % 32 == 0)
#pragma unroll
        for (int v = 0; v < 8; ++v) obase[(size_t)v * 64] = acc[v];
    } else {
#pragma unroll
        for (int v = 0; v < 8; ++v) {
            if (mbase + v < nrows) obase[(size_t)v * 64] = acc[v];
        }
    }
}

// ---------------------------------------------------------------------------
// per-node prep: attention logits al_src/al_dst, init max=-inf, sum=0, acc=0
// ---------------------------------------------------------------------------
template <int H, int C>
__global__ void prep_kernel(const float* __restrict__ h, const float* __restrict__ a_src,
                            const float* __restrict__ a_dst, float* __restrict__ alS,
                            float* __restrict__ alD, float* __restrict__ mmax,
                            float* __restrict__ ssum, float* __restrict__ acc, int n) {
    const int i = blockIdx.x * blockDim.x + threadIdx.x;
    if (i >= n) return;
    const float* __restrict__ hr = h + (size_t)i * 64;
    float as[H], ad[H];
#pragma unroll
    for (int hh = 0; hh < H; ++hh) { as[hh] = 0.f; ad[hh] = 0.f; }
#pragma unroll
    for (int j = 0; j < 64; ++j) {
        const int hh = j / C;            // compile-time per unrolled iteration
        const float v = hr[j];
        as[hh] += v * a_src[j];
        ad[hh] += v * a_dst[j];
    }
#pragma unroll
    for (int hh = 0; hh < H; ++hh) {
        alS[i * H + hh] = as[hh];
        alD[i * H + hh] = ad[hh];
        mmax[i * H + hh] = -__builtin_inff();
        ssum[i * H + hh] = 0.f;
    }
    float4* __restrict__ ac = (float4*)(acc + (size_t)i * 64);
#pragma unroll
    for (int j = 0; j < 16; ++j) ac[j] = make_float4(0.f, 0.f, 0.f, 0.f);
}

// ---------------------------------------------------------------------------
// edge pass 1: segment max of leaky_relu(al_src[s]+al_dst[d]); thread = (edge,head)
// ---------------------------------------------------------------------------
template <int H>
__global__ void edge_max_kernel(const int* __restrict__ ei, const float* __restrict__ alS,
                                const float* __restrict__ alD, float* __restrict__ mmax,
                                int E, int etot) {
    const int t = blockIdx.x * blockDim.x + threadIdx.x;
    if (t >= etot * H) return;
    const int e = t / H;
    const int hh = t - e * H;
    int s, d;
    if (e < E) { s = ei[e]; d = ei[E + e]; } else { s = d = e - E; }
    float ev = alS[s * H + hh] + alD[d * H + hh];
    ev = (ev > 0.f) ? ev : LRELU_SLOPE * ev;
    atomicMaxF32(&mmax[d * H + hh], ev);
}

// ---------------------------------------------------------------------------
// edge pass 2 (fused): ee = exp(e - max); ssum[d] += ee; acc[d] += ee*h[s]
// 16 threads per edge, 4 channels each (coalesced 256B gather of h[src])
// ---------------------------------------------------------------------------
template <int H, int C>
__global__ void edge_agg_kernel(const int* __restrict__ ei, const float* __restrict__ h,
                                const float* __restrict__ alS, const float* __restrict__ alD,
                                const float* __restrict__ mmax, float* __restrict__ ssum,
                                float* __restrict__ acc, int E, int etot) {
    const long long t = (long long)blockIdx.x * blockDim.x + threadIdx.x;
    const int e = (int)(t >> 4);
    const int r = (int)(t & 15);
    if (e >= etot) return;
    int s, d;
    if (e < E) { s = ei[e]; d = ei[E + e]; } else { s = d = e - E; }
    const int c0 = r * 4;
    const int hh = c0 / C;
    float ev = alS[s * H + hh] + alD[d * H + hh];
    ev = (ev > 0.f) ? ev : LRELU_SLOPE * ev;
    const float ee = __expf(ev - mmax[d * H + hh]);
    if ((c0 & (C - 1)) == 0) atomicAdd(&ssum[d * H + hh], ee);
    const float4 hv = *(const float4*)(h + (size_t)s * 64 + c0);
    float* __restrict__ a = acc + (size_t)d * 64 + c0;
    atomicAdd(a + 0, ee * hv.x);
    atomicAdd(a + 1, ee * hv.y);
    atomicAdd(a + 2, ee * hv.z);
    atomicAdd(a + 3, ee * hv.w);
}

// ---------------------------------------------------------------------------
// per-node epilogue (in place): acc/s + bias -> batchnorm -> optional ELU
// ---------------------------------------------------------------------------
template <int H, int C>
__global__ void finish_kernel(float* __restrict__ acc, const float* __restrict__ ssum,
                              const float* __restrict__ bias, const float* __restrict__ gamma,
                              const float* __restrict__ beta, const float* __restrict__ mean,
                              const float* __restrict__ var, int n, int do_elu) {
    const int i = blockIdx.x * blockDim.x + threadIdx.x;
    if (i >= n) return;
    float* __restrict__ row = acc + (size_t)i * 64;
    float sinv[H];
#pragma unroll
    for (int hh = 0; hh < H; ++hh) sinv[hh] = 1.f / (ssum[i * H + hh] + 1e-16f);
#pragma unroll
    for (int j = 0; j < 64; ++j) {
        const int hh = j / C;
        float v = row[j] * sinv[hh] + bias[j];
        v = (v - mean[j]) * rsqrtf(var[j] + BN_EPS) * gamma[j] + beta[j];
        if (do_elu) v = (v > 0.f) ? v : (expm1f(v));
        row[j] = v;
    }
}

// ---------------------------------------------------------------------------
// mean pooling over nodes -> pooled[64] (block-partial sums + atomics)
// ---------------------------------------------------------------------------
__global__ void zero64_kernel(float* __restrict__ p) { p[threadIdx.x] = 0.f; }

__global__ void pool_kernel(const float* __restrict__ h, float* __restrict__ pooled, int n) {
    __shared__ float sh[256];
    const int j = threadIdx.x & 63;
    const int g = threadIdx.x >> 6;
    float sum = 0.f;
    for (int node = blockIdx.x * 4 + g; node < n; node += gridDim.x * 4)
        sum += h[(size_t)node * 64 + j];
    sh[threadIdx.x] = sum;
    __syncthreads();
    if (g == 0) {
        const float v = sh[j] + sh[64 + j] + sh[128 + j] + sh[192 + j];
        atomicAdd(&pooled[j], v);
    }
}

// ---------------------------------------------------------------------------
// final MLP heads (1 block, 64 threads): class logits [3] + rul [1] -> d_out[4]
// ---------------------------------------------------------------------------
__global__ void head_kernel(const float* __restrict__ pooled,
                            const float* __restrict__ cw1, const float* __restrict__ cb1,
                            const float* __restrict__ cw2, const float* __restrict__ cb2,
                            const float* __restrict__ rw1, const float* __restrict__ rb1,
                            const float* __restrict__ rw2, const float* __restrict__ rb2,
                            float* __restrict__ out, float invn) {
    __shared__ float p[64];
    __shared__ float hid[64];
    const int t = threadIdx.x;
    p[t] = pooled[t] * invn;
    __syncthreads();
    {
        const float* wrow = (t < 32) ? (cw1 + t * 64) : (rw1 + (t - 32) * 64);
        float s = (t < 32) ? cb1[t] : rb1[t - 32];
        for (int j = 0; j < 64; ++j) s += wrow[j] * p[j];
        hid[t] = fmaxf(s, 0.f);
    }
    __syncthreads();
    if (t < 3) {
        float s = cb2[t];
        for (int i = 0; i < 32; ++i) s += cw2[t * 32 + i] * hid[i];
        out[t] = s;
    } else if (t == 3) {
        float s = rb2[0];
        for (int i = 0; i < 32; ++i) s += rw2[i] * hid[32 + i];
        out[3] = s;
    }
}

// ---------------------------------------------------------------------------
extern "C" void kernel_launch(void* const* d_in, const int* in_sizes, int n_in,
                              void* d_out, int out_size, void* d_ws, size_t ws_size,
                              hipStream_t stream) {
    const float* x      = (const float*)d_in[0];
    const int*   ei     = (const int*)  d_in[1];
    const float* W1     = (const float*)d_in[2];
    const float* a1s    = (const float*)d_in[3];
    const float* a1d    = (const float*)d_in[4];
    const float* b1     = (const float*)d_in[5];
    const float* W2     = (const float*)d_in[6];
    const float* a2s    = (const float*)d_in[7];
    const float* a2d    = (const float*)d_in[8];
    const float* b2     = (const float*)d_in[9];
    const float* W3     = (const float*)d_in[10];
    const float* a3s    = (const float*)d_in[11];
    const float* a3d    = (const float*)d_in[12];
    const float* b3     = (const float*)d_in[13];
    const float* bng    = (const float*)d_in[14];
    const float* bnb    = (const float*)d_in[15];
    const float* bnm    = (const float*)d_in[16];
    const float* bnv    = (const float*)d_in[17];
    const float* cw1    = (const float*)d_in[18];
    const float* cb1    = (const float*)d_in[19];
    const float* cw2    = (const float*)d_in[20];
    const float* cb2    = (const float*)d_in[21];
    const float* rw1    = (const float*)d_in[22];
    const float* rb1    = (const float*)d_in[23];
    const float* rw2    = (const float*)d_in[24];
    const float* rb2    = (const float*)d_in[25];

    const int N = in_sizes[0] / 5;
    const int E = in_sizes[1] / 2;
    const int etot = E + N;

    float* ws    = (float*)d_ws;
    float* h_tmp = ws;                              // N*64
    float* acc   = h_tmp + (size_t)N * 64;          // N*64 (becomes layer output)
    float* alS   = acc   + (size_t)N * 64;          // N*4
    float* alD   = alS   + (size_t)N * 4;           // N*4
    float* mmax  = alD   + (size_t)N * 4;           // N*4
    float* ssum  = mmax  + (size_t)N * 4;           // N*4
    float* pooled= ssum  + (size_t)N * 4;           // 64

    const int nodeBlocks = (N + 255) / 256;
    const int gemmBlocks = (N + 31) / 32;
    const int emaxBlocks4 = (etot * 4 + 255) / 256;
    const int emaxBlocks1 = (etot + 255) / 256;
    const int eaggBlocks  = (int)(((long long)etot * 16 + 255) / 256);

    // ---- layer 1: fin=5, H=4, C=16, BN[0] + ELU ----
    gemm64_wmma<5><<<gemmBlocks, 256, 0, stream>>>(x, W1, h_tmp, N);
    prep_kernel<4, 16><<<nodeBlocks, 256, 0, stream>>>(h_tmp, a1s, a1d, alS, alD, mmax, ssum, acc, N);
    edge_max_kernel<4><<<emaxBlocks4, 256, 0, stream>>>(ei, alS, alD, mmax, E, etot);
    edge_agg_kernel<4, 16><<<eaggBlocks, 256, 0, stream>>>(ei, h_tmp, alS, alD, mmax, ssum, acc, E, etot);
    finish_kernel<4, 16><<<nodeBlocks, 256, 0, stream>>>(acc, ssum, b1, bng, bnb, bnm, bnv, N, 1);

    // ---- layer 2: fin=64, H=4, C=16, BN[1] + ELU ----
    gemm64_wmma<64><<<gemmBlocks, 256, 0, stream>>>(acc, W2, h_tmp, N);
    prep_kernel<4, 16><<<nodeBlocks, 256, 0, stream>>>(h_tmp, a2s, a2d, alS, alD, mmax, ssum, acc, N);
    edge_max_kernel<4><<<emaxBlocks4, 256, 0, stream>>>(ei, alS, alD, mmax, E, etot);
    edge_agg_kernel<4, 16><<<eaggBlocks, 256, 0, stream>>>(ei, h_tmp, alS, alD, mmax, ssum, acc, E, etot);
    finish_kernel<4, 16><<<nodeBlocks, 256, 0, stream>>>(acc, ssum, b2, bng + 64, bnb + 64, bnm + 64, bnv + 64, N, 1);

    // ---- layer 3: fin=64, H=1, C=64, BN[2], no ELU ----
    gemm64_wmma<64><<<gemmBlocks, 256, 0, stream>>>(acc, W3, h_tmp, N);
    prep_kernel<1, 64><<<nodeBlocks, 256, 0, stream>>>(h_tmp, a3s, a3d, alS, alD, mmax, ssum, acc, N);
    edge_max_kernel<1><<<emaxBlocks1, 256, 0, stream>>>(ei, alS, alD, mmax, E, etot);
    edge_agg_kernel<1, 64><<<eaggBlocks, 256, 0, stream>>>(ei, h_tmp, alS, alD, mmax, ssum, acc, E, etot);
    finish_kernel<1, 64><<<nodeBlocks, 256, 0, stream>>>(acc, ssum, b3, bng + 128, bnb + 128, bnm + 128, bnv + 128, N, 0);

    // ---- mean pool + heads ----
    zero64_kernel<<<1, 64, 0, stream>>>(pooled);
    pool_kernel<<<400, 256, 0, stream>>>(acc, pooled, N);
    head_kernel<<<1, 64, 0, stream>>>(pooled, cw1, cb1, cw2, cb2, rw1, rb1, rw2, rb2,
                                      (float*)d_out, 1.0f / (float)N);
}